// ScorePredictor_30983894073819
// MI455X (gfx1250) — compile-verified
//
#include <hip/hip_runtime.h>
#include <hip/hip_bf16.h>

typedef __attribute__((ext_vector_type(2))) float v2f;
typedef __attribute__((ext_vector_type(8))) float v8f;

#define D_FEAT 128
#define WAVES_PER_BLOCK 8
#define EDGES_PER_WAVE 16

// One wave handles 16 edges.
// A-matrix (16x4 f32 per step): row i = x[src[e0+i]], lane i (and i+16) supplies it.
// B-matrix (4x16 f32 per step): col j = x[dst[e0+j]], lane j (and j+16) supplies it.
// 32 accumulating V_WMMA_F32_16X16X4_F32 steps cover K=128; the 16 edge
// dot-products are the diagonal of the resulting 16x16 f32 accumulator.
__global__ __launch_bounds__(256) void edge_dot_wmma_f32(
    const float* __restrict__ x,
    const int*   __restrict__ src,
    const int*   __restrict__ dst,
    float*       __restrict__ out,
    int n_edges)
{
    const int lane    = threadIdx.x & 31;
    const int wave    = threadIdx.x >> 5;
    const long long waveGlobal = (long long)blockIdx.x * WAVES_PER_BLOCK + wave;
    const long long e0 = waveGlobal * EDGES_PER_WAVE;
    if (e0 >= n_edges) return;               // wave-uniform: EXEC stays all-1s

    // Edge this lane feeds (lanes L and L+16 feed the same edge, different K).
    const int ei = lane & 15;
    long long e = e0 + ei;
    if (e >= n_edges) e = n_edges - 1;       // clamp instead of predicating

    const int s = src[e];
    const int d = dst[e];

    // Column offset within each K=4 step: lanes 0-15 -> K={0,1}, 16-31 -> K={2,3}.
    const int colOff = (lane >> 4) << 1;
    const float* __restrict__ ps = x + (long long)s * D_FEAT + colOff;
    const float* __restrict__ pd = x + (long long)d * D_FEAT + colOff;

    v8f c = {};
#pragma unroll
    for (int k0 = 0; k0 < D_FEAT; k0 += 4) {
        v2f a = *(const v2f*)(ps + k0);      // global_load_b64, imm offset
        v2f b = *(const v2f*)(pd + k0);      // global_load_b64, imm offset
        // 8 args: (neg_a, A, neg_b, B, c_mod, C, reuse_a, reuse_b)
        c = __builtin_amdgcn_wmma_f32_16x16x4_f32(
                false, a, false, b, (short)0, c, false, false);
    }

    // Diagonal D[i][i]:
    //   i = 0..7  : lane i,     VGPR i        -> lanes 0..7,   c[lane]
    //   i = 8..15 : lane i+16,  VGPR i-8      -> lanes 24..31, c[lane-24]
    const int idx = lane & 7;
    float val = c[0];
    if (idx == 1) val = c[1];
    if (idx == 2) val = c[2];
    if (idx == 3) val = c[3];
    if (idx == 4) val = c[4];
    if (idx == 5) val = c[5];
    if (idx == 6) val = c[6];
    if (idx == 7) val = c[7];

    const bool writer = (lane < 8) || (lane >= 24);
    const long long eo = e0 + ((lane < 8) ? lane : (lane - 16));
    if (writer && eo < n_edges) out[eo] = val;
}

extern "C" void kernel_launch(void* const* d_in, const int* in_sizes, int n_in,
                              void* d_out, int out_size, void* d_ws, size_t ws_size,
                              hipStream_t stream) {
    const float* x   = (const float*)d_in[0];
    const int*   src = (const int*)d_in[1];
    const int*   dst = (const int*)d_in[2];
    float*       out = (float*)d_out;

    const int n_edges = in_sizes[1];
    const int n_waves = (n_edges + EDGES_PER_WAVE - 1) / EDGES_PER_WAVE;
    const int blocks  = (n_waves + WAVES_PER_BLOCK - 1) / WAVES_PER_BLOCK;

    edge_dot_wmma_f32<<<blocks, WAVES_PER_BLOCK * 32, 0, stream>>>(
        x, src, dst, out, n_edges);
}